// TTTWrapper_40656160424179
// MI455X (gfx1250) — compile-verified
//
#include <hip/hip_runtime.h>
#include <math.h>

// ---------------------------------------------------------------------------
// MI455X / gfx1250 TTT layer.  All GEMMs run on V_WMMA_F32_16X16X4_F32
// (fp32 in/out, wave32).  Fragment layouts per CDNA5 ISA 7.12.2.
// ---------------------------------------------------------------------------

typedef float v2f __attribute__((ext_vector_type(2)));
typedef float v8f __attribute__((ext_vector_type(8)));

#define WMMA4(a, b, c) \
  __builtin_amdgcn_wmma_f32_16x16x4_f32(false, (a), false, (b), (short)0, (c), false, false)

constexpr int Bn = 16, Nn = 1024, Cn = 768, Hn = 12, Dn = 64;
constexpr int Mn = Bn * Nn;           // 16384 token rows
constexpr float LN_EPS = 1e-6f;

// ---------------------------------------------------------------------------
// K1: qkv = x @ Wqkv^T + bias, scattered into q/k/v buffers [B,H,N,D].
// Each wave: one 16(M) x 64(N) tile; 64-wide column tile maps to a unique
// (q|k|v, head) pair since 2304 = 3*12*64.
// ---------------------------------------------------------------------------
__global__ __launch_bounds__(256) void k1_qkv_gemm(
    const float* __restrict__ x, const float* __restrict__ w,   // w: [2304,768]
    const float* __restrict__ qbias, const float* __restrict__ vbias,
    float* __restrict__ qo, float* __restrict__ ko, float* __restrict__ vo)
{
  const int wid  = blockIdx.x * 8 + (threadIdx.x >> 5);
  const int lane = threadIdx.x & 31;
  const int row  = lane & 15, hi = lane >> 4;
  const int ct = wid % 36;                 // column tile (64 wide)
  const int mt = wid / 36;                 // row tile (16 tall)
  const int n0 = ct * 64, m0 = mt * 16;

  const float* arow = x + (size_t)(m0 + row) * Cn;
  const float* bp0  = w + (size_t)(n0 +  0 + row) * Cn;
  const float* bp1  = w + (size_t)(n0 + 16 + row) * Cn;
  const float* bp2  = w + (size_t)(n0 + 32 + row) * Cn;
  const float* bp3  = w + (size_t)(n0 + 48 + row) * Cn;

  v8f c0 = {}, c1 = {}, c2 = {}, c3 = {};
#pragma unroll 4
  for (int k0 = 0; k0 < Cn; k0 += 4) {
    const int kk = k0 + 2 * hi;
    v2f a  = *(const v2f*)(arow + kk);      // A[row][kk], A[row][kk+1]
    v2f b0 = *(const v2f*)(bp0 + kk);       // B[kk][col] = W[col][kk]
    v2f b1 = *(const v2f*)(bp1 + kk);
    v2f b2 = *(const v2f*)(bp2 + kk);
    v2f b3 = *(const v2f*)(bp3 + kk);
    c0 = WMMA4(a, b0, c0);
    c1 = WMMA4(a, b1, c1);
    c2 = WMMA4(a, b2, c2);
    c3 = WMMA4(a, b3, c3);
  }

  const int which = ct / Hn;               // 0=q 1=k 2=v
  const int h     = ct % Hn;
  float* outp = (which == 0) ? qo : (which == 1 ? ko : vo);
  v8f cacc[4] = {c0, c1, c2, c3};
#pragma unroll
  for (int f = 0; f < 4; ++f) {
    const int d = 16 * f + row;
    float bias = (which == 0) ? qbias[h * Dn + d]
               : (which == 2) ? vbias[h * Dn + d] : 0.0f;
#pragma unroll
    for (int r = 0; r < 8; ++r) {
      const int m = m0 + r + 8 * hi;       // token row
      const int b = m >> 10, n = m & 1023;
      outp[((size_t)(b * Hn + h) * Nn + n) * Dn + d] = cacc[f][r] + bias;
    }
  }
}

// ---------------------------------------------------------------------------
// K2: eta[b,h,n] = sigmoid(x . lr_weight[h] + lr_bias[h]) / D.
// One wave per token; x row cached in registers (24 floats/lane).
// ---------------------------------------------------------------------------
__global__ __launch_bounds__(256) void k2_eta(
    const float* __restrict__ x, const float* __restrict__ lrw,
    const float* __restrict__ lrb, float* __restrict__ eta)
{
  const int token = blockIdx.x * 8 + (threadIdx.x >> 5);
  const int lane  = threadIdx.x & 31;
  const float* xp = x + (size_t)token * Cn;
  float xr[24];
#pragma unroll
  for (int i = 0; i < 24; ++i) xr[i] = xp[lane + 32 * i];
  const int b = token >> 10, n = token & 1023;
  for (int h = 0; h < Hn; ++h) {
    const float* wp = lrw + (size_t)h * Cn;
    float s = 0.0f;
#pragma unroll
    for (int i = 0; i < 24; ++i) s += xr[i] * wp[lane + 32 * i];
#pragma unroll
    for (int m = 16; m >= 1; m >>= 1) s += __shfl_xor(s, m, 32);
    if (lane == 0) {
      float t  = s + lrb[h];
      float sg = 1.0f / (1.0f + __expf(-t));
      eta[(size_t)(b * Hn + h) * Nn + n] = sg * (1.0f / (float)Dn);
    }
  }
}

// ---------------------------------------------------------------------------
// K3: per (b,h): z = k@W1 + b1 ; LN fwd ; dy=(y-(v-k))*eta/N ; LN bwd -> dz.
// Wave = 16 rows x 64 cols (whole D).  Row stats via 16-lane shfl_xor.
// ---------------------------------------------------------------------------
__global__ __launch_bounds__(256) void k3_inner_grad(
    const float* __restrict__ kbuf, const float* __restrict__ vbuf,
    const float* __restrict__ eta,  const float* __restrict__ W1,
    const float* __restrict__ b1,   const float* __restrict__ gamma,
    const float* __restrict__ beta, float* __restrict__ dz)
{
  const int bh = blockIdx.x >> 3;
  const int rt = blockIdx.x & 7;
  const int wv = threadIdx.x >> 5;
  const int lane = threadIdx.x & 31;
  const int row = lane & 15, hi = lane >> 4;
  const int h  = bh % Hn;
  const int m0 = rt * 128 + wv * 16;

  const float* kp = kbuf + (size_t)bh * Nn * Dn;
  const float* vp = vbuf + (size_t)bh * Nn * Dn;
  const float* W  = W1 + (size_t)h * Dn * Dn;      // [d_in][d_out] row-major
  const float* arow = kp + (size_t)(m0 + row) * Dn;

  v8f c[4] = {};
#pragma unroll 4
  for (int k0 = 0; k0 < Dn; k0 += 4) {
    const int kk = k0 + 2 * hi;
    v2f a = *(const v2f*)(arow + kk);
#pragma unroll
    for (int f = 0; f < 4; ++f) {
      v2f bb;
      bb.x = W[(size_t)kk * Dn + 16 * f + row];
      bb.y = W[(size_t)(kk + 1) * Dn + 16 * f + row];
      c[f] = WMMA4(a, bb, c[f]);
    }
  }

  float b1v[4], gm[4], bt[4];
#pragma unroll
  for (int f = 0; f < 4; ++f) {
    const int d = 16 * f + row;
    b1v[f] = b1[h * Dn + d];
    gm[f]  = gamma[h * Dn + d];
    bt[f]  = beta[h * Dn + d];
  }
#pragma unroll
  for (int f = 0; f < 4; ++f)
#pragma unroll
    for (int r = 0; r < 8; ++r) c[f][r] += b1v[f];

  float* dzp = dz + (size_t)bh * Nn * Dn;
#pragma unroll
  for (int r = 0; r < 8; ++r) {
    const int n = m0 + r + 8 * hi;
    // LN forward stats over D=64: 4 regs x 16 lanes (same half).
    float s1 = 0.0f, s2 = 0.0f;
#pragma unroll
    for (int f = 0; f < 4; ++f) { float z = c[f][r]; s1 += z; s2 += z * z; }
#pragma unroll
    for (int m = 8; m >= 1; m >>= 1) { s1 += __shfl_xor(s1, m, 32); s2 += __shfl_xor(s2, m, 32); }
    const float mu   = s1 * (1.0f / Dn);
    const float var  = s2 * (1.0f / Dn) - mu * mu;
    const float rstd = rsqrtf(var + LN_EPS);
    const float etan = eta[(size_t)bh * Nn + n] * (1.0f / (float)Nn);

    float xh[4], g[4];
    float mg = 0.0f, mgx = 0.0f;
#pragma unroll
    for (int f = 0; f < 4; ++f) {
      const int d = 16 * f + row;
      xh[f] = (c[f][r] - mu) * rstd;
      const float y   = gm[f] * xh[f] + bt[f];
      const float tgt = vp[(size_t)n * Dn + d] - kp[(size_t)n * Dn + d];
      const float dy  = (y - tgt) * etan;
      g[f] = gm[f] * dy;
      mg  += g[f];
      mgx += g[f] * xh[f];
    }
#pragma unroll
    for (int m = 8; m >= 1; m >>= 1) { mg += __shfl_xor(mg, m, 32); mgx += __shfl_xor(mgx, m, 32); }
    mg  *= (1.0f / Dn);
    mgx *= (1.0f / Dn);
#pragma unroll
    for (int f = 0; f < 4; ++f) {
      const int d = 16 * f + row;
      dzp[(size_t)n * Dn + d] = rstd * (g[f] - mg - xh[f] * mgx);
    }
  }
}

// ---------------------------------------------------------------------------
// K4: per (b,h): gW = k^T @ dz (64x64, K=1024), gb = colsum(dz).
// Writes W_new = W1 - gW, b_new = b1 - gb.  4 waves, each 16x64 strip.
// ---------------------------------------------------------------------------
__global__ __launch_bounds__(128) void k4_grad_update(
    const float* __restrict__ kbuf, const float* __restrict__ dz,
    const float* __restrict__ W1,   const float* __restrict__ b1,
    float* __restrict__ Wn, float* __restrict__ bn)
{
  const int bh = blockIdx.x;
  const int h  = bh % Hn;
  const int wv = threadIdx.x >> 5;
  const int lane = threadIdx.x & 31;
  const int row = lane & 15, hi = lane >> 4;
  const int m0 = wv * 16;

  const float* kp  = kbuf + (size_t)bh * Nn * Dn;
  const float* dzp = dz   + (size_t)bh * Nn * Dn;

  v8f c[4] = {};
#pragma unroll 2
  for (int kk0 = 0; kk0 < Nn; kk0 += 4) {
    const int kk = kk0 + 2 * hi;
    v2f a;                                        // A = k^T : A[m][kk] = k[kk][m]
    a.x = kp[(size_t)kk * Dn + m0 + row];
    a.y = kp[(size_t)(kk + 1) * Dn + m0 + row];
#pragma unroll
    for (int f = 0; f < 4; ++f) {
      v2f bb;                                     // B = dz : B[kk][n]
      bb.x = dzp[(size_t)kk * Dn + 16 * f + row];
      bb.y = dzp[(size_t)(kk + 1) * Dn + 16 * f + row];
      c[f] = WMMA4(a, bb, c[f]);
    }
  }

  const float* W = W1 + (size_t)h * Dn * Dn;
  float* Wo = Wn + (size_t)bh * Dn * Dn;
#pragma unroll
  for (int f = 0; f < 4; ++f) {
    const int n = 16 * f + row;
#pragma unroll
    for (int r = 0; r < 8; ++r) {
      const int m = m0 + r + 8 * hi;
      Wo[(size_t)m * Dn + n] = W[(size_t)m * Dn + n] - c[f][r];
    }
  }

  // gb = colsum(dz); b_new = b1 - gb
  __shared__ float sm[128];
  const int n = threadIdx.x & 63, part = threadIdx.x >> 6;
  float s = 0.0f;
  for (int t = part; t < Nn; t += 2) s += dzp[(size_t)t * Dn + n];
  sm[threadIdx.x] = s;
  __syncthreads();
  if (threadIdx.x < 64)
    bn[(size_t)bh * Dn + n] = b1[h * Dn + n] - (sm[threadIdx.x] + sm[threadIdx.x + 64]);
}

// ---------------------------------------------------------------------------
// K5: per (b,h): y = LN(q @ W_new + b_new); attn[b,n, h*64+d] = y + q.
// ---------------------------------------------------------------------------
__global__ __launch_bounds__(256) void k5_ttt_out(
    const float* __restrict__ qbuf, const float* __restrict__ Wn,
    const float* __restrict__ bn,   const float* __restrict__ gamma,
    const float* __restrict__ beta, float* __restrict__ attn)
{
  const int bh = blockIdx.x >> 3;
  const int rt = blockIdx.x & 7;
  const int wv = threadIdx.x >> 5;
  const int lane = threadIdx.x & 31;
  const int row = lane & 15, hi = lane >> 4;
  const int h = bh % Hn, b = bh / Hn;
  const int m0 = rt * 128 + wv * 16;

  const float* qp = qbuf + (size_t)bh * Nn * Dn;
  const float* W  = Wn + (size_t)bh * Dn * Dn;
  const float* arow = qp + (size_t)(m0 + row) * Dn;

  v8f c[4] = {};
#pragma unroll 4
  for (int k0 = 0; k0 < Dn; k0 += 4) {
    const int kk = k0 + 2 * hi;
    v2f a = *(const v2f*)(arow + kk);
#pragma unroll
    for (int f = 0; f < 4; ++f) {
      v2f bb;
      bb.x = W[(size_t)kk * Dn + 16 * f + row];
      bb.y = W[(size_t)(kk + 1) * Dn + 16 * f + row];
      c[f] = WMMA4(a, bb, c[f]);
    }
  }

  float bv[4], gm[4], bt[4];
#pragma unroll
  for (int f = 0; f < 4; ++f) {
    const int d = 16 * f + row;
    bv[f] = bn[(size_t)bh * Dn + d];
    gm[f] = gamma[h * Dn + d];
    bt[f] = beta[h * Dn + d];
  }
#pragma unroll
  for (int f = 0; f < 4; ++f)
#pragma unroll
    for (int r = 0; r < 8; ++r) c[f][r] += bv[f];

#pragma unroll
  for (int r = 0; r < 8; ++r) {
    const int n = m0 + r + 8 * hi;
    float s1 = 0.0f, s2 = 0.0f;
#pragma unroll
    for (int f = 0; f < 4; ++f) { float z = c[f][r]; s1 += z; s2 += z * z; }
#pragma unroll
    for (int m = 8; m >= 1; m >>= 1) { s1 += __shfl_xor(s1, m, 32); s2 += __shfl_xor(s2, m, 32); }
    const float mu   = s1 * (1.0f / Dn);
    const float var  = s2 * (1.0f / Dn) - mu * mu;
    const float rstd = rsqrtf(var + LN_EPS);
#pragma unroll
    for (int f = 0; f < 4; ++f) {
      const int d = 16 * f + row;
      const float y  = gm[f] * (c[f][r] - mu) * rstd + bt[f];
      const float qv = qp[(size_t)n * Dn + d];
      attn[((size_t)(b * Nn + n)) * Cn + h * Dn + d] = y + qv;
    }
  }
}

// ---------------------------------------------------------------------------
// K6: out = attn @ proj_weight^T + proj_bias   (16384x768 @ 768x768)
// ---------------------------------------------------------------------------
__global__ __launch_bounds__(256) void k6_proj_gemm(
    const float* __restrict__ attn, const float* __restrict__ w,
    const float* __restrict__ bias, float* __restrict__ out)
{
  const int wid  = blockIdx.x * 8 + (threadIdx.x >> 5);
  const int lane = threadIdx.x & 31;
  const int row  = lane & 15, hi = lane >> 4;
  const int ct = wid % 12, mt = wid / 12;
  const int n0 = ct * 64, m0 = mt * 16;

  const float* arow = attn + (size_t)(m0 + row) * Cn;
  const float* bp0 = w + (size_t)(n0 +  0 + row) * Cn;
  const float* bp1 = w + (size_t)(n0 + 16 + row) * Cn;
  const float* bp2 = w + (size_t)(n0 + 32 + row) * Cn;
  const float* bp3 = w + (size_t)(n0 + 48 + row) * Cn;

  v8f c0 = {}, c1 = {}, c2 = {}, c3 = {};
#pragma unroll 4
  for (int k0 = 0; k0 < Cn; k0 += 4) {
    const int kk = k0 + 2 * hi;
    v2f a  = *(const v2f*)(arow + kk);
    v2f b0 = *(const v2f*)(bp0 + kk);
    v2f b1 = *(const v2f*)(bp1 + kk);
    v2f b2 = *(const v2f*)(bp2 + kk);
    v2f b3 = *(const v2f*)(bp3 + kk);
    c0 = WMMA4(a, b0, c0);
    c1 = WMMA4(a, b1, c1);
    c2 = WMMA4(a, b2, c2);
    c3 = WMMA4(a, b3, c3);
  }

  v8f cacc[4] = {c0, c1, c2, c3};
#pragma unroll
  for (int f = 0; f < 4; ++f) {
    const int col = n0 + 16 * f + row;
    const float bv = bias[col];
#pragma unroll
    for (int r = 0; r < 8; ++r) {
      const int m = m0 + r + 8 * hi;
      out[(size_t)m * Cn + col] = cacc[f][r] + bv;
    }
  }
}

// ---------------------------------------------------------------------------
// Host-side launcher.  Workspace layout (floats):
//   qbuf | kbuf | vbuf (reused as attn) | dzbuf | eta | W_new | b_new
// ---------------------------------------------------------------------------
extern "C" void kernel_launch(void* const* d_in, const int* in_sizes, int n_in,
                              void* d_out, int out_size, void* d_ws, size_t ws_size,
                              hipStream_t stream)
{
  const float* x    = (const float*)d_in[0];
  const float* qkvw = (const float*)d_in[1];
  const float* qb   = (const float*)d_in[2];
  const float* vb   = (const float*)d_in[3];
  const float* pw   = (const float*)d_in[4];
  const float* pb   = (const float*)d_in[5];
  const float* lrw  = (const float*)d_in[6];
  const float* lrb  = (const float*)d_in[7];
  const float* gam  = (const float*)d_in[8];
  const float* bet  = (const float*)d_in[9];
  const float* W1   = (const float*)d_in[10];
  const float* b1   = (const float*)d_in[11];
  float* out = (float*)d_out;

  const size_t per = (size_t)Mn * Cn;           // 12,582,912 floats
  float* ws    = (float*)d_ws;
  float* qbuf  = ws;
  float* kbuf  = qbuf + per;
  float* vbuf  = kbuf + per;
  float* dzbuf = vbuf + per;
  float* eta   = dzbuf + per;
  float* Wn    = eta + (size_t)Bn * Hn * Nn;    // 196,608 floats
  float* bn    = Wn + (size_t)Bn * Hn * Dn * Dn;// 786,432 floats
  float* attn  = vbuf;                          // v dead after K3

  // K1: 16384/16 row-tiles * 36 col-tiles = 36864 waves / 8 per block
  k1_qkv_gemm<<<4608, 256, 0, stream>>>(x, qkvw, qb, vb, qbuf, kbuf, vbuf);
  // K2: 16384 tokens, 8 waves/block
  k2_eta<<<2048, 256, 0, stream>>>(x, lrw, lrb, eta);
  // K3: 192 heads * 8 row-tiles-of-128
  k3_inner_grad<<<1536, 256, 0, stream>>>(kbuf, vbuf, eta, W1, b1, gam, bet, dzbuf);
  // K4: one block per (b,h)
  k4_grad_update<<<192, 128, 0, stream>>>(kbuf, dzbuf, W1, b1, Wn, bn);
  // K5: same shape as K3
  k5_ttt_out<<<1536, 256, 0, stream>>>(qbuf, Wn, bn, gam, bet, attn);
  // K6: 1024 row-tiles * 12 col-tiles / 8
  k6_proj_gemm<<<1536, 256, 0, stream>>>(attn, pw, pb, out);
}